// BasicModel_76390288327245
// MI455X (gfx1250) — compile-verified
//
#include <hip/hip_runtime.h>

typedef __attribute__((ext_vector_type(2))) float v2f;
typedef __attribute__((ext_vector_type(8))) float v8f;

#define BATCH  4096
#define DIM    32
#define NPROD  100000
#define KTOP   100
#define TILE_M 16
#define CW     256   // candidates per chunk in scores kernel (2 tiles per wave)

__device__ __forceinline__ v8f wmma_f32_4(v2f a, v2f b, v8f c) {
  // D = A(16x4 f32) x B(4x16 f32) + C(16x16 f32)
  return __builtin_amdgcn_wmma_f32_16x16x4_f32(false, a, false, b, (short)0, c,
                                               false, false);
}

// ---------------------------------------------------------------------------
// Kernel 1: embedding gathers -> d_out (user_emb, product_emb)
// ---------------------------------------------------------------------------
__global__ void gather_emb_kernel(const int* __restrict__ uid,
                                  const int* __restrict__ pid,
                                  const float* __restrict__ utab,
                                  const float* __restrict__ ptab,
                                  float* __restrict__ outU,
                                  float* __restrict__ outP) {
  int i = blockIdx.x * blockDim.x + threadIdx.x;
  if (i >= BATCH * DIM) return;
  int r = i >> 5, c = i & 31;
  outU[i] = utab[(size_t)uid[r] * DIM + c];
  outP[i] = ptab[(size_t)pid[r] * DIM + c];
}

// ---------------------------------------------------------------------------
// Kernel 2: fused MLP, one 16-row tile per workgroup, all layers in LDS.
// relu(X@W1+b1) -> relu(@W2+b2) -> @W3+b3. X gathered in-kernel.
// ---------------------------------------------------------------------------
__global__ void mlp_kernel(const int* __restrict__ uid, const int* __restrict__ pid,
                           const float* __restrict__ utab, const float* __restrict__ ptab,
                           const float* __restrict__ W1, const float* __restrict__ b1,
                           const float* __restrict__ W2, const float* __restrict__ b2,
                           const float* __restrict__ W3, const float* __restrict__ b3,
                           float* __restrict__ outR) {
  __shared__ float X[TILE_M][64];
  __shared__ float H1[TILE_M][256];
  __shared__ float H2[TILE_M][128];

  int tid  = threadIdx.x;           // 256 threads = 8 waves
  int tile = blockIdx.x;

  // Gather concat([user_emb, product_emb]) tile: 16x64 floats.
  for (int e = 0; e < 4; ++e) {
    int idx = tid * 4 + e;          // 0..1023
    int r = idx >> 6, c = idx & 63;
    int g = tile * TILE_M + r;
    float v = (c < DIM) ? utab[(size_t)uid[g] * DIM + c]
                        : ptab[(size_t)pid[g] * DIM + (c - DIM)];
    X[r][c] = v;
  }
  __syncthreads();

  int lane = tid & 31;
  int wave = tid >> 5;
  int m    = lane & 15;
  int koff = (lane < 16) ? 0 : 2;

  // Layer 1: (16x64)@(64x256), each wave owns two 16-col tiles, 16 k-steps.
  for (int s = 0; s < 2; ++s) {
    int n = (wave * 2 + s) * 16 + (lane & 15);
    v8f acc = {};
    for (int t = 0; t < 16; ++t) {
      v2f a, b;
      a.x = X[m][4 * t + koff];
      a.y = X[m][4 * t + koff + 1];
      b.x = W1[(size_t)(4 * t + koff) * 256 + n];
      b.y = W1[(size_t)(4 * t + koff + 1) * 256 + n];
      acc = wmma_f32_4(a, b, acc);
    }
    float bias = b1[n];
    for (int v = 0; v < 8; ++v) {
      int mr = v + ((lane < 16) ? 0 : 8);
      float h = acc[v] + bias;
      H1[mr][n] = h > 0.f ? h : 0.f;
    }
  }
  __syncthreads();

  // Layer 2: (16x256)@(256x128), each wave one 16-col tile, 64 k-steps.
  {
    int n = wave * 16 + (lane & 15);
    v8f acc = {};
    for (int t = 0; t < 64; ++t) {
      v2f a, b;
      a.x = H1[m][4 * t + koff];
      a.y = H1[m][4 * t + koff + 1];
      b.x = W2[(size_t)(4 * t + koff) * 128 + n];
      b.y = W2[(size_t)(4 * t + koff + 1) * 128 + n];
      acc = wmma_f32_4(a, b, acc);
    }
    float bias = b2[n];
    for (int v = 0; v < 8; ++v) {
      int mr = v + ((lane < 16) ? 0 : 8);
      float h = acc[v] + bias;
      H2[mr][n] = h > 0.f ? h : 0.f;
    }
  }
  __syncthreads();

  // Layer 3: (16x128)@(128x1)
  if (tid < TILE_M) {
    float s = b3[0];
    for (int k = 0; k < 128; ++k) s += H2[tid][k] * W3[k];
    outR[tile * TILE_M + tid] = s;
  }
}

// ---------------------------------------------------------------------------
// Kernel 3: scores = user_emb @ candidates.T via fp32 WMMA + streaming top-100.
// One workgroup (8 waves) per 16-user tile; chunks of 256 candidates
// (2 x 16x16 WMMA tiles per wave per chunk -> 16 WMMAs between barriers).
// ---------------------------------------------------------------------------
__global__ void scores_topk_kernel(const int* __restrict__ uid,
                                   const float* __restrict__ utab,
                                   const float* __restrict__ ptab,
                                   float* __restrict__ outS,
                                   float* __restrict__ outI) {
  __shared__ float Uemb[TILE_M][DIM];       //  2 KB
  __shared__ float chunk[TILE_M][CW];       // 16 KB
  __shared__ float stageS[TILE_M][CW];      // 16 KB
  __shared__ int   stageI[TILE_M][CW];      // 16 KB
  __shared__ int   stageCnt[TILE_M];
  __shared__ float listS[TILE_M][KTOP];     // 6.4 KB
  __shared__ int   listI[TILE_M][KTOP];     // 6.4 KB
  __shared__ int   listCnt[TILE_M];
  __shared__ float thresh[TILE_M];

  int tid  = threadIdx.x;           // 256 threads = 8 waves
  int tile = blockIdx.x;

  for (int e = 0; e < 2; ++e) {
    int idx = tid * 2 + e;          // 0..511
    int r = idx >> 5, c = idx & 31;
    Uemb[r][c] = utab[(size_t)uid[tile * TILE_M + r] * DIM + c];
  }
  if (tid < TILE_M) {
    listCnt[tid]  = 0;
    stageCnt[tid] = 0;
    thresh[tid]   = -__builtin_inff();
  }
  __syncthreads();

  int lane = tid & 31;
  int wave = tid >> 5;
  int mm   = lane & 15;
  int koff = (lane < 16) ? 0 : 2;

  // A-fragments (user tile) loaded once, resident in VGPRs for all chunks.
  v2f aF[8];
  for (int t = 0; t < 8; ++t) {
    aF[t].x = Uemb[mm][4 * t + koff];
    aF[t].y = Uemb[mm][4 * t + koff + 1];
  }

  int u   = tid >> 4;               // user within tile (0..15)
  int sub = tid & 15;               // helper thread within user group

  for (int c0 = 0; c0 < NPROD; c0 += CW) {
    // --- GEMM: this wave -> two 16-candidate tiles of the chunk ---
    for (int s = 0; s < 2; ++s) {
      int col = (wave * 2 + s) * 16 + mm;               // 0..255 within chunk
      int n   = c0 + col;
      int nc  = n < NPROD ? n : NPROD - 1;              // clamp OOB loads
      const float* crow = ptab + (size_t)(nc + 1) * DIM;  // candidates = table[1:]
      v8f acc = {};
      for (int t = 0; t < 8; ++t) {                     // K = 32 in steps of 4
        v2f b;
        b.x = crow[4 * t + koff];
        b.y = crow[4 * t + koff + 1];
        acc = wmma_f32_4(aF[t], b, acc);
      }
      for (int v = 0; v < 8; ++v) {
        int mr = v + ((lane < 16) ? 0 : 8);
        chunk[mr][col] = acc[v];
      }
    }
    __syncthreads();

    // --- filter chunk against running threshold (16 cols per helper) ---
    float th = (listCnt[u] >= KTOP) ? thresh[u] : -__builtin_inff();
    for (int j = 0; j < CW / 16; ++j) {
      int col  = sub * (CW / 16) + j;
      int cand = c0 + col;
      if (cand < NPROD) {
        float s = chunk[u][col];
        if (s > th) {
          int pos = atomicAdd(&stageCnt[u], 1);
          stageS[u][pos] = s;
          stageI[u][pos] = cand;
        }
      }
    }
    __syncthreads();

    // --- one thread per user: sorted insertion of staged survivors ---
    if (sub == 0) {
      int cnt = listCnt[u];
      int ns  = stageCnt[u];
      for (int i = 0; i < ns; ++i) {
        float s  = stageS[u][i];
        int   id = stageI[u][i];
        if (cnt < KTOP) {
          int p = cnt;
          while (p > 0 && listS[u][p - 1] < s) {
            listS[u][p] = listS[u][p - 1];
            listI[u][p] = listI[u][p - 1];
            --p;
          }
          listS[u][p] = s; listI[u][p] = id; ++cnt;
        } else if (s > listS[u][KTOP - 1]) {
          int p = KTOP - 1;
          while (p > 0 && listS[u][p - 1] < s) {
            listS[u][p] = listS[u][p - 1];
            listI[u][p] = listI[u][p - 1];
            --p;
          }
          listS[u][p] = s; listI[u][p] = id;
        }
      }
      listCnt[u] = cnt;
      if (cnt >= KTOP) thresh[u] = listS[u][KTOP - 1];
      stageCnt[u] = 0;
    }
    __syncthreads();
  }

  // --- write top-k (scores + indices-as-float) ---
  int r = tile * TILE_M + u;
  for (int e = sub; e < KTOP; e += 16) {
    outS[(size_t)r * KTOP + e] = listS[u][e];
    outI[(size_t)r * KTOP + e] = (float)listI[u][e];
  }
}

// ---------------------------------------------------------------------------
extern "C" void kernel_launch(void* const* d_in, const int* in_sizes, int n_in,
                              void* d_out, int out_size, void* d_ws, size_t ws_size,
                              hipStream_t stream) {
  const int*   uid  = (const int*)d_in[0];
  const int*   pid  = (const int*)d_in[1];
  const float* utab = (const float*)d_in[2];
  const float* ptab = (const float*)d_in[3];
  const float* W1   = (const float*)d_in[4];
  const float* b1   = (const float*)d_in[5];
  const float* W2   = (const float*)d_in[6];
  const float* b2   = (const float*)d_in[7];
  const float* W3   = (const float*)d_in[8];
  const float* b3   = (const float*)d_in[9];

  float* out  = (float*)d_out;
  float* outU = out;                              // 4096*32
  float* outP = outU + BATCH * DIM;               // 4096*32
  float* outR = outP + BATCH * DIM;               // 4096*1
  float* outS = outR + BATCH;                     // 4096*100
  float* outI = outS + BATCH * KTOP;              // 4096*100 (idx as float)

  gather_emb_kernel<<<(BATCH * DIM + 255) / 256, 256, 0, stream>>>(
      uid, pid, utab, ptab, outU, outP);
  mlp_kernel<<<BATCH / TILE_M, 256, 0, stream>>>(
      uid, pid, utab, ptab, W1, b1, W2, b2, W3, b3, outR);
  scores_topk_kernel<<<BATCH / TILE_M, 256, 0, stream>>>(
      uid, utab, ptab, outS, outI);
}